// QuantizerLUT_13580686590013
// MI455X (gfx1250) — compile-verified
//
#include <hip/hip_runtime.h>

#ifndef __has_builtin
#define __has_builtin(x) 0
#endif

#define GS   128                 // elements per quantization group
#define LEV  16                  // levels per group
#define TPB  256                 // threads per block (8 wave32)
#define EPT  4                   // elements per thread (one float4)
#define EPB  (TPB * EPT)         // 1024 elements per block
#define GPB  (EPB / GS)          // 8 groups per block

// The async-to-LDS builtin's real signature (from the round-1 diagnostic):
//   void __builtin_amdgcn_global_load_async_to_lds_b128(
//       int4 __attribute__((address_space(1)))* src_global,
//       int4 __attribute__((address_space(3)))* dst_lds,
//       int imm_offset, int cpol);
typedef int v4i __attribute__((vector_size(16)));
typedef __attribute__((address_space(1))) v4i Gv4i;
typedef __attribute__((address_space(3))) v4i Lv4i;

__global__ __launch_bounds__(TPB)
void quantlut_ste_kernel(const float* __restrict__ x,
                         const float* __restrict__ levels,
                         float* __restrict__ out,
                         long long n, long long lev_count) {
  __shared__ __align__(16) float s_lev[GPB * LEV];   // 8 groups x 16 levels = 512 B

  const int tid = threadIdx.x;
  const long long blk_base = (long long)blockIdx.x * EPB;
  const long long base     = blk_base + (long long)tid * EPT;
  const long long g_base   = blk_base / GS;          // first group of this block

  // ---- 1) issue this thread's x load early; it overlaps the staging barrier
  float4 xv = make_float4(0.f, 0.f, 0.f, 0.f);
  const bool full = (base + EPT) <= n;
  if (full) xv = *(const float4*)(x + base);

  // ---- 2) stage the block's 8x16 level table into LDS via the CDNA5 async path
#if __has_builtin(__builtin_amdgcn_global_load_async_to_lds_b128)
  if (tid < 32) {
    const long long loff = g_base * LEV + (long long)tid * 4;
    if (loff + 4 <= lev_count) {
      // one wave32 b128 async op moves all 512 bytes: lane t -> s_lev[4t..4t+3]
      __builtin_amdgcn_global_load_async_to_lds_b128(
          (Gv4i*)(levels + loff), (Lv4i*)&s_lev[tid * 4], 0, 0);
    }
#if __has_builtin(__builtin_amdgcn_s_wait_asynccnt)
    __builtin_amdgcn_s_wait_asynccnt(0);
#else
    asm volatile("s_wait_asynccnt 0x0" ::: "memory");
#endif
  }
#else
  if (tid < GPB * LEV) {
    const long long loff = g_base * LEV + tid;
    if (loff < lev_count) s_lev[tid] = levels[loff];
  }
#endif
  __syncthreads();

  // ---- 3) pull my group's 16 levels from LDS (wave-uniform address -> broadcast)
  const int lg = tid >> 5;                      // local group 0..7 (uniform per wave)
  float lv[LEV];
  {
    const float4* s4 = (const float4*)&s_lev[lg * LEV];
    float4 a = s4[0], b = s4[1], c = s4[2], d = s4[3];
    lv[0]  = a.x; lv[1]  = a.y; lv[2]  = a.z; lv[3]  = a.w;
    lv[4]  = b.x; lv[5]  = b.y; lv[6]  = b.z; lv[7]  = b.w;
    lv[8]  = c.x; lv[9]  = c.y; lv[10] = c.z; lv[11] = c.w;
    lv[12] = d.x; lv[13] = d.y; lv[14] = d.z; lv[15] = d.w;
  }
  float bd[LEV - 1];
#pragma unroll
  for (int j = 0; j < LEV - 1; ++j) bd[j] = 0.5f * (lv[j + 1] + lv[j]);

  // branchless bucketize: idx = #borders strictly below x; q = lv[idx]
  auto quant = [&](float xc) -> float {
    float q = lv[0];
#pragma unroll
    for (int j = 1; j < LEV; ++j) q = (xc > bd[j - 1]) ? lv[j] : q;
    return (q - xc) + xc;                       // STE forward, same eval order as ref
  };

  // ---- 4) compute + coalesced b128 store
  if (full) {
    float4 o;
    o.x = quant(xv.x);
    o.y = quant(xv.y);
    o.z = quant(xv.z);
    o.w = quant(xv.w);
    *(float4*)(out + base) = o;
  } else {
#pragma unroll
    for (int k = 0; k < EPT; ++k) {
      const long long i = base + k;
      if (i < n) {
        float xc = x[i];
        out[i] = quant(xc);
      }
    }
  }
}

extern "C" void kernel_launch(void* const* d_in, const int* in_sizes, int n_in,
                              void* d_out, int out_size, void* d_ws, size_t ws_size,
                              hipStream_t stream) {
  const float* x      = (const float*)d_in[0];
  const float* levels = (const float*)d_in[1];
  float* out          = (float*)d_out;
  const long long n         = (long long)in_sizes[0];   // 4096*4096
  const long long lev_count = (long long)in_sizes[1];   // 131072*16
  const long long blocks = (n + EPB - 1) / EPB;         // 16384 blocks
  hipLaunchKernelGGL(quantlut_ste_kernel, dim3((unsigned)blocks), dim3(TPB), 0, stream,
                     x, levels, out, n, lev_count);
}